// Pytorch_FeatureAttention_70282844832166
// MI455X (gfx1250) — compile-verified
//
#include <hip/hip_runtime.h>
#include <hip/hip_bf16.h>

typedef __attribute__((ext_vector_type(2))) float v2f;
typedef __attribute__((ext_vector_type(8))) float v8f;

#define B_   32
#define L_   512
#define C_   128
#define HID_ 256
#define H_   4
#define D_   64
#define NL_  3
#define NEG_ 0.2f
#define EPS_ 1e-5f

// ---------------------------------------------------------------------------
// Fully-constant-shape batched f32 GEMM on V_WMMA_F32_16X16X4_F32.
// 32x32 of D per wave (2x2 tiles of 16x16, 4 v8f accumulators), K step 4.
// All dims/strides are template parameters -> address math folds into
// immediate offsets. TRANSA=0 A-fragments are fetched as one b64 load
// (adjacent + 8B aligned); B fragments are LDB apart (two b32 loads).
// A frag (16x4 f32): v0 = A[m0+r, k0+2*half], v1 = A[m0+r, k0+1+2*half]
// B frag (4x16 f32): v0 = B[k0+2*half, n0+r], v1 = B[k0+1+2*half, n0+r]
// C/D frag:          acc[i] = D[m0 + i + 8*half, n0 + r]
// Grid: x = (M/32)*(N/32)/4 (exact), y = batch (outer*INNER + inner).
// ---------------------------------------------------------------------------
template <int M, int N, int K, int LDA, int LDB, int LDD, int INNER,
          long SAO, long SAI, long SBO, long SBI, long SDO, long SDI,
          long SBIASI, bool TRANSA, bool TRANSD, bool HASBIAS>
__global__ __launch_bounds__(128) void wmma_gemm_f32(
    const float* __restrict__ A, const float* __restrict__ Bm,
    const float* __restrict__ bias, float* __restrict__ D)
{
  const int lane = threadIdx.x & 31;
  const int wave = threadIdx.x >> 5;
  const int half = (lane >> 4) & 1;
  const int r    = lane & 15;

  constexpr int tilesN = N >> 5;
  const int tile = blockIdx.x * 4 + wave;       // grid divides exactly
  const int tm = tile / tilesN;
  const int tn = tile % tilesN;

  const int batch = blockIdx.y;
  const int bo = batch / INNER;
  const int bi = batch % INNER;
  A    += bo * SAO + bi * SAI;
  Bm   += bo * SBO + bi * SBI;
  D    += bo * SDO + bi * SDI;
  if (HASBIAS) bias += bi * SBIASI;

  const int m0 = tm << 5;
  const int n0 = tn << 5;

  constexpr int stepA = TRANSA ? 4 * LDA : 4;   // per K-step advance
  constexpr int stepB = 4 * LDB;
  int offA0 = TRANSA ? (2 * half) * LDA + (m0 + r)
                     : (m0 + r) * LDA + 2 * half;
  int offA1 = offA0 + (TRANSA ? 16 : 16 * LDA);
  int offB0 = (2 * half) * LDB + (n0 + r);
  int offB1 = offB0 + 16;

  v8f acc00 = {0.f,0.f,0.f,0.f,0.f,0.f,0.f,0.f};
  v8f acc01 = acc00, acc10 = acc00, acc11 = acc00;

#pragma unroll 4
  for (int k0 = 0; k0 < K; k0 += 4) {
    v2f a0, a1, b0, b1;
    if (TRANSA) {
      a0.x = A[offA0]; a0.y = A[offA0 + LDA];
      a1.x = A[offA1]; a1.y = A[offA1 + LDA];
    } else {
      a0 = *(const v2f*)(A + offA0);            // global_load_b64
      a1 = *(const v2f*)(A + offA1);            // global_load_b64
    }
    b0.x = Bm[offB0]; b0.y = Bm[offB0 + LDB];
    b1.x = Bm[offB1]; b1.y = Bm[offB1 + LDB];
    offA0 += stepA; offA1 += stepA; offB0 += stepB; offB1 += stepB;

    acc00 = __builtin_amdgcn_wmma_f32_16x16x4_f32(false, a0, false, b0, (short)0, acc00, false, false);
    acc01 = __builtin_amdgcn_wmma_f32_16x16x4_f32(false, a0, false, b1, (short)0, acc01, false, false);
    acc10 = __builtin_amdgcn_wmma_f32_16x16x4_f32(false, a1, false, b0, (short)0, acc10, false, false);
    acc11 = __builtin_amdgcn_wmma_f32_16x16x4_f32(false, a1, false, b1, (short)0, acc11, false, false);
  }

  const int col0 = n0 + r;
  const int col1 = col0 + 16;
  const float bv0 = HASBIAS ? bias[col0] : 0.f;
  const float bv1 = HASBIAS ? bias[col1] : 0.f;
#pragma unroll
  for (int i = 0; i < 8; ++i) {
    const int row0 = m0 + i + 8 * half;
    const int row1 = row0 + 16;
    if (TRANSD) {
      D[(long)col0 * LDD + row0] = acc00[i] + bv0;
      D[(long)col1 * LDD + row0] = acc01[i] + bv1;
      D[(long)col0 * LDD + row1] = acc10[i] + bv0;
      D[(long)col1 * LDD + row1] = acc11[i] + bv1;
    } else {
      D[(long)row0 * LDD + col0] = acc00[i] + bv0;
      D[(long)row0 * LDD + col1] = acc01[i] + bv1;
      D[(long)row1 * LDD + col0] = acc10[i] + bv0;
      D[(long)row1 * LDD + col1] = acc11[i] + bv1;
    }
  }
}

// ---------------------------------------------------------------------------
// GATv2 scores + softmax over sources for one (b, h):
//   alpha[b,h,t,s] = softmax_s( sum_d att[h,d] * lrelu(gl[b,t,h,d] + gr[b,s,h,d]) )
// gr head tile (128x64 = 32KB) cached in LDS; 128 threads, thread = s.
// ---------------------------------------------------------------------------
__global__ __launch_bounds__(128) void gat_scores_softmax(
    const float* __restrict__ gl, const float* __restrict__ gr,
    const float* __restrict__ att, float* __restrict__ alpha)
{
  __shared__ float sGr[C_ * D_];   // 32 KB
  __shared__ float sAtt[D_];
  __shared__ float sRow[D_];
  __shared__ float red[4];

  const int tid = threadIdx.x;
  const int h = blockIdx.x;
  const int b = blockIdx.y;
  const float* glb = gl + (long)b * C_ * HID_ + h * D_;
  const float* grb = gr + (long)b * C_ * HID_ + h * D_;

  for (int idx = tid; idx < C_ * D_; idx += 128) {
    const int s = idx >> 6, d = idx & 63;
    sGr[idx] = grb[(long)s * HID_ + d];
  }
  if (tid < D_) sAtt[tid] = att[h * D_ + tid];
  __syncthreads();

  const int lane = tid & 31;
  const int wv = tid >> 5;
  float* arow = alpha + (long)(b * H_ + h) * C_ * C_;

  for (int t = 0; t < C_; ++t) {
    if (tid < D_) sRow[tid] = glb[(long)t * HID_ + tid];
    __syncthreads();

    const int s = tid;
    float sc = 0.f;
    const float* g = &sGr[s * D_];
#pragma unroll 8
    for (int d = 0; d < D_; ++d) {
      float v = sRow[d] + g[d];
      v = v > 0.f ? v : NEG_ * v;
      sc = fmaf(sAtt[d], v, sc);
    }
    // block max over 128 sources (4 waves)
    float m = sc;
    for (int off = 16; off > 0; off >>= 1) m = fmaxf(m, __shfl_xor(m, off, 32));
    if (lane == 0) red[wv] = m;
    __syncthreads();
    m = fmaxf(fmaxf(red[0], red[1]), fmaxf(red[2], red[3]));
    __syncthreads();
    const float e = __expf(sc - m);
    float ssum = e;
    for (int off = 16; off > 0; off >>= 1) ssum += __shfl_xor(ssum, off, 32);
    if (lane == 0) red[wv] = ssum;
    __syncthreads();
    ssum = red[0] + red[1] + red[2] + red[3];
    arow[(long)t * C_ + s] = e / ssum;
    __syncthreads();   // before sRow / red are rewritten next iteration
  }
}

// ---------------------------------------------------------------------------
// h = LayerNorm(h + elu(gout)) * g + b        (one (b,c) row per 256-thr block)
// ---------------------------------------------------------------------------
__global__ __launch_bounds__(256) void post_elu_residual_ln(
    float* __restrict__ h, const float* __restrict__ gout,
    const float* __restrict__ g, const float* __restrict__ bb)
{
  __shared__ float red[8];
  const int tid = threadIdx.x;
  const long base = (long)blockIdx.x * HID_;

  float x = gout[base + tid];
  x = x > 0.f ? x : (__expf(x) - 1.f);           // elu (alpha = 1)
  const float v = h[base + tid] + x;             // residual

  const int lane = tid & 31, wv = tid >> 5;
  float s = v;
  for (int off = 16; off > 0; off >>= 1) s += __shfl_xor(s, off, 32);
  if (lane == 0) red[wv] = s;
  __syncthreads();
  s = 0.f;
#pragma unroll
  for (int i = 0; i < 8; ++i) s += red[i];
  const float mu = s * (1.f / HID_);
  __syncthreads();

  const float dmu = v - mu;
  float q = dmu * dmu;
  for (int off = 16; off > 0; off >>= 1) q += __shfl_xor(q, off, 32);
  if (lane == 0) red[wv] = q;
  __syncthreads();
  q = 0.f;
#pragma unroll
  for (int i = 0; i < 8; ++i) q += red[i];
  const float inv = rsqrtf(q * (1.f / HID_) + EPS_);

  h[base + tid] = dmu * inv * g[tid] + bb[tid];
}

// ---------------------------------------------------------------------------
extern "C" void kernel_launch(void* const* d_in, const int* in_sizes, int n_in,
                              void* d_out, int out_size, void* d_ws, size_t ws_size,
                              hipStream_t stream) {
  (void)in_sizes; (void)n_in; (void)out_size; (void)ws_size;
  const float* x      = (const float*)d_in[0];   // (B,L,C)
  const float* W_emb  = (const float*)d_in[1];   // (L,HID)
  const float* b_emb  = (const float*)d_in[2];   // (HID)
  const float* Wl     = (const float*)d_in[3];   // (NL,HID,HID)
  const float* Wr     = (const float*)d_in[4];   // (NL,HID,HID)
  const float* att    = (const float*)d_in[5];   // (NL,H,D)
  const float* bias_g = (const float*)d_in[6];   // (NL,HID)
  const float* ln_g   = (const float*)d_in[7];   // (NL,HID)
  const float* ln_b   = (const float*)d_in[8];   // (NL,HID)
  const float* W_proj = (const float*)d_in[9];   // (HID,L)
  const float* b_proj = (const float*)d_in[10];  // (L)

  float* out   = (float*)d_out;                  // (B,L,C)
  float* alpha = out + (long)B_ * L_ * C_;       // (B,H,C,C) — last layer's survives

  float* ws = (float*)d_ws;
  const long hwN = (long)B_ * C_ * HID_;         // 1,048,576 floats
  float* h    = ws;
  float* gl   = ws + hwN;
  float* gr   = ws + 2 * hwN;
  float* gout = ws + 3 * hwN;                    // total 16 MB workspace

  // h[b] = x[b]^T(128x512) @ W_emb(512x256) + b_emb
  wmma_gemm_f32<C_, HID_, L_, C_, HID_, HID_, 1,
                (long)L_ * C_, 0, 0, 0, (long)C_ * HID_, 0, 0,
                true, false, true>
      <<<dim3((C_ / 32) * (HID_ / 32) / 4, B_), 128, 0, stream>>>(
          x, W_emb, b_emb, h);

  for (int layer = 0; layer < NL_; ++layer) {
    const float* Wl_i  = Wl + (long)layer * HID_ * HID_;
    const float* Wr_i  = Wr + (long)layer * HID_ * HID_;
    const float* att_i = att + (long)layer * H_ * D_;
    const float* bg_i  = bias_g + (long)layer * HID_;

    dim3 g1((C_ / 32) * (HID_ / 32) / 4, B_);
    wmma_gemm_f32<C_, HID_, HID_, HID_, HID_, HID_, 1,
                  (long)C_ * HID_, 0, 0, 0, (long)C_ * HID_, 0, 0,
                  false, false, false>
        <<<g1, 128, 0, stream>>>(h, Wl_i, nullptr, gl);
    wmma_gemm_f32<C_, HID_, HID_, HID_, HID_, HID_, 1,
                  (long)C_ * HID_, 0, 0, 0, (long)C_ * HID_, 0, 0,
                  false, false, false>
        <<<g1, 128, 0, stream>>>(h, Wr_i, nullptr, gr);

    gat_scores_softmax<<<dim3(H_, B_), 128, 0, stream>>>(gl, gr, att_i, alpha);

    // gout[b, t, h*64+d] = sum_s alpha[b,h,t,s] * gr[b,s,h*64+d] + bias_g[h*64+d]
    wmma_gemm_f32<C_, D_, C_, C_, HID_, HID_, H_,
                  (long)H_ * C_ * C_, (long)C_ * C_,
                  (long)C_ * HID_, D_, (long)C_ * HID_, D_, D_,
                  false, false, true>
        <<<dim3((C_ / 32) * (D_ / 32) / 4, B_ * H_), 128, 0, stream>>>(
            alpha, gr, bg_i, gout);

    post_elu_residual_ln<<<B_ * C_, 256, 0, stream>>>(
        h, gout, ln_g + (long)layer * HID_, ln_b + (long)layer * HID_);
  }

  // out[b,l,c] = (h[b](128x256) @ W_proj(256x512))[c,l] + b_proj[l]  (transposed store)
  wmma_gemm_f32<C_, L_, HID_, HID_, L_, C_, 1,
                (long)C_ * HID_, 0, 0, 0, (long)L_ * C_, 0, 0,
                false, true, true>
      <<<dim3((C_ / 32) * (L_ / 32) / 4, B_), 128, 0, stream>>>(
          h, W_proj, b_proj, out);
}